// GCAPCN_5892695130351
// MI455X (gfx1250) — compile-verified
//
#include <hip/hip_runtime.h>

#define B_  32
#define N_  512
#define BN_ 16384          // B*N
#define D_  384            // F0_DIM
#define ND_ 128            // N_DIM

typedef float v2f __attribute__((ext_vector_type(2)));
typedef float v8f __attribute__((ext_vector_type(8)));
typedef int   v4i __attribute__((vector_size(16)));   // 16-byte payload type for async LDS DMA

// workspace layout (float offsets)
#define OFF_STATS 0
#define OFF_BN    16
#define OFF_X     1024
#define OFF_L     65536
#define OFF_F0    (OFF_L  + (size_t)B_ * N_ * N_)
#define OFF_T1    (OFF_F0 + (size_t)BN_ * D_)
#define OFF_T2    (OFF_T1 + (size_t)BN_ * D_)
#define OFF_F1    (OFF_T2 + (size_t)BN_ * D_)

__device__ inline float lrelu(float v) { return v > 0.0f ? v : 0.01f * v; }
__device__ inline float applyPow(float v, int p) { return p == 1 ? v : (p == 2 ? v * v : v * v * v); }

// ---------------- async global->LDS staging (CDNA5) ----------------

#if __has_builtin(__builtin_amdgcn_global_load_async_to_lds_b128) && \
    __has_builtin(__builtin_amdgcn_s_wait_asynccnt)
#define HAVE_ASYNC_LDS 1
#else
#define HAVE_ASYNC_LDS 0
#endif

__device__ inline void copy_b128_to_lds(const float* __restrict__ g, float* __restrict__ l) {
#if HAVE_ASYNC_LDS
  __builtin_amdgcn_global_load_async_to_lds_b128(
      (__attribute__((address_space(1))) v4i*)g,
      (__attribute__((address_space(3))) v4i*)l, 0, 0);
#else
  *(float4*)l = *(const float4*)g;   // global_load_b128 + ds_store_b128
#endif
}
__device__ inline void async_copies_wait() {
#if HAVE_ASYNC_LDS
  __builtin_amdgcn_s_wait_asynccnt(0);
#endif
}

// ---------------- WMMA fragment helpers (fp32 16x16x4, ISA 7.12.2 layouts) ----------------

// A 16x4: lanes 0-15 rows with K={k0,k0+1}; lanes 16-31 K={k0+2,k0+3}
__device__ inline v2f load_a_frag(const float* __restrict__ A, int lda, int m0, int k0, int lane) {
  int row = m0 + (lane & 15);
  int kb  = k0 + ((lane >> 4) << 1);
  const float* p = A + (size_t)row * lda + kb;
  v2f r; r.x = p[0]; r.y = p[1];
  return r;
}
// B 4x16: row-striped across lanes, half-wave split on K
__device__ inline v2f load_b_frag(const float* __restrict__ Bm, int ldb, int k0, int n0, int lane) {
  int col = n0 + (lane & 15);
  int kb  = k0 + ((lane >> 4) << 1);
  v2f r;
  r.x = Bm[(size_t)kb * ldb + col];
  r.y = Bm[(size_t)(kb + 1) * ldb + col];
  return r;
}

#define WMMA4(acc, a, b) \
  (acc) = __builtin_amdgcn_wmma_f32_16x16x4_f32(false, (a), false, (b), (short)0, (acc), false, false)

// ---------------- setup kernels ----------------

__global__ void k_init(const float* __restrict__ deadline, float* __restrict__ stats) {
  __shared__ float sm[256];
  int tid = threadIdx.x;
  float m = -1e30f;
  for (int i = tid; i < BN_; i += 256) m = fmaxf(m, deadline[i]);
  sm[tid] = m; __syncthreads();
  for (int s = 128; s > 0; s >>= 1) { if (tid < s) sm[tid] = fmaxf(sm[tid], sm[tid + s]); __syncthreads(); }
  if (tid == 0) { stats[0] = sm[0]; ((unsigned int*)stats)[1] = 0x7F800000u; }
}

__global__ void k_build_XF0(const float* __restrict__ loc, const float* __restrict__ deadline,
                            const float* __restrict__ Winit, const float* __restrict__ binit,
                            const float* __restrict__ stats,
                            float* __restrict__ X, float* __restrict__ F0) {
  int row = blockIdx.x;                       // b*512 + n
  float x0 = loc[row * 2 + 0], x1 = loc[row * 2 + 1];
  float x2 = deadline[row] / stats[0];
  if (threadIdx.x == 0) { X[row * 3 + 0] = x0; X[row * 3 + 1] = x1; X[row * 3 + 2] = x2; }
  for (int c = threadIdx.x; c < D_; c += blockDim.x)
    F0[(size_t)row * D_ + c] = x0 * Winit[c] + x1 * Winit[D_ + c] + x2 * Winit[2 * D_ + c] + binit[c];
}

__global__ void k_d2min(const float* __restrict__ X, unsigned int* __restrict__ statsu) {
  __shared__ unsigned int sm[256];
  size_t idx = (size_t)blockIdx.x * 256 + threadIdx.x;  // < 32*512*512
  int b   = (int)(idx >> 18);
  int rem = (int)(idx & 262143);
  int i = rem >> 9, j = rem & 511;
  unsigned int local = 0x7F800000u;
  if (i != j) {
    const float* Xi = X + ((size_t)b * N_ + i) * 3;
    const float* Xj = X + ((size_t)b * N_ + j) * 3;
    float d0 = Xi[0] - Xj[0], d1 = Xi[1] - Xj[1], dc = Xi[2] - Xj[2];
    local = __float_as_uint(d0 * d0 + d1 * d1 + dc * dc);
  }
  sm[threadIdx.x] = local; __syncthreads();
  for (int s = 128; s > 0; s >>= 1) {
    if (threadIdx.x < s) sm[threadIdx.x] = sm[threadIdx.x] < sm[threadIdx.x + s] ? sm[threadIdx.x] : sm[threadIdx.x + s];
    __syncthreads();
  }
  if (threadIdx.x == 0) atomicMin(&statsu[1], sm[0]);
}

__global__ void k_build_L(const float* __restrict__ X, const float* __restrict__ stats,
                          float* __restrict__ L) {
  __shared__ float sm[512];
  int blk = blockIdx.x;        // b*512 + i
  int j = threadIdx.x;
  int b = blk >> 9, i = blk & 511;
  float d2min = __uint_as_float(((const unsigned int*)stats)[1]);
  float sq = sqrtf(d2min);     // A_norm = sqrt(d2min)/sqrt(d2)
  const float* Xi = X + (size_t)blk * 3;
  const float* Xj = X + ((size_t)b * N_ + j) * 3;
  float d0 = Xi[0] - Xj[0], d1 = Xi[1] - Xj[1], dc = Xi[2] - Xj[2];
  float d2 = d0 * d0 + d1 * d1 + dc * dc;
  float a = (i == j) ? 0.0f : sq * rsqrtf(d2);
  sm[j] = a; __syncthreads();
  for (int s = 256; s > 0; s >>= 1) { if (j < s) sm[j] += sm[j + s]; __syncthreads(); }
  float deg = sm[0] - 1.0f;
  L[(size_t)blk * N_ + j] = (i == j) ? deg : -a;
}

__global__ void k_pow(const float* __restrict__ F0, float* __restrict__ out, int p) {
  size_t i = (size_t)blockIdx.x * 256 + threadIdx.x;
  out[i] = applyPow(F0[i], p);
}

// ---------------- LDS-staged WMMA GEMM: C(b) = L(b) @ Fin(b) ----------------
// Block tile 128(M) x 64(N), K-chunk 32, double-buffered async global->LDS staging.
// 8 waves as 4x2, wave tile 32x32 (2x2 wmma tiles).

#define KC   32
#define APAD 36   // row stride (floats) for A tile in LDS: 16B-aligned, conflict-free
#define BPAD 72   // row stride (floats) for B tile in LDS: half-waves hit disjoint banks

__global__ __launch_bounds__(256) void k_gemm_L(const float* __restrict__ Lm,
                                                const float* __restrict__ Fin,
                                                float* __restrict__ Out) {
  __shared__ float sA[2][128 * APAD];
  __shared__ float sB[2][KC * BPAD];

  int b = blockIdx.z;
  const float* A  = Lm  + (size_t)b * N_ * N_ + (size_t)blockIdx.x * 128 * N_;  // 128-row panel
  const float* Bm = Fin + (size_t)b * N_ * D_;
  float* C = Out + (size_t)b * N_ * D_;
  int nBase = blockIdx.y * 64;

  int tid  = threadIdx.x;
  int lane = tid & 31, wave = tid >> 5;
  int wm = (wave & 3) * 32;            // wave tile M offset within block tile
  int wn = (wave >> 2) * 32;           // wave tile N offset within block tile

  // staging coordinates
  int rowA = tid >> 3;                 // 0..31, +32 stride (4 iters covers 128 rows)
  int segA = (tid & 7) * 4;            // 16B segment within 32-float A row
  int rowB = tid >> 4;                 // 0..15, +16 stride (2 iters covers 32 rows)
  int segB = (tid & 15) * 4;           // 16B segment within 64-float B row

  // stage chunk 0
  {
    float* dA = sA[0]; float* dB = sB[0];
    for (int r = rowA; r < 128; r += 32)
      copy_b128_to_lds(A + (size_t)r * N_ + segA, dA + r * APAD + segA);
    for (int r = rowB; r < KC; r += 16)
      copy_b128_to_lds(Bm + (size_t)r * D_ + nBase + segB, dB + r * BPAD + segB);
  }
  async_copies_wait();
  __syncthreads();

  v8f acc[2][2] = {};
  for (int chunk = 0; chunk < N_ / KC; ++chunk) {
    int cur = chunk & 1;
    // kick off DMA for the next chunk into the other buffer (overlaps with WMMA below)
    if (chunk + 1 < N_ / KC) {
      int kc = (chunk + 1) * KC;
      float* dA = sA[cur ^ 1]; float* dB = sB[cur ^ 1];
      for (int r = rowA; r < 128; r += 32)
        copy_b128_to_lds(A + (size_t)r * N_ + kc + segA, dA + r * APAD + segA);
      for (int r = rowB; r < KC; r += 16)
        copy_b128_to_lds(Bm + (size_t)(kc + r) * D_ + nBase + segB, dB + r * BPAD + segB);
    }
    const float* cA = sA[cur];
    const float* cB = sB[cur];
#pragma unroll
    for (int k0 = 0; k0 < KC; k0 += 4) {
      int kb = k0 + ((lane >> 4) << 1);
      int ra = wm + (lane & 15);
      v2f a0; a0.x = cA[ra * APAD + kb];        a0.y = cA[ra * APAD + kb + 1];
      v2f a1; a1.x = cA[(ra + 16) * APAD + kb]; a1.y = cA[(ra + 16) * APAD + kb + 1];
      int cb = wn + (lane & 15);
      v2f b0; b0.x = cB[kb * BPAD + cb];        b0.y = cB[(kb + 1) * BPAD + cb];
      v2f b1; b1.x = cB[kb * BPAD + cb + 16];   b1.y = cB[(kb + 1) * BPAD + cb + 16];
      WMMA4(acc[0][0], a0, b0); WMMA4(acc[0][1], a0, b1);
      WMMA4(acc[1][0], a1, b0); WMMA4(acc[1][1], a1, b1);
    }
    async_copies_wait();
    __syncthreads();
  }

  int mGlob = blockIdx.x * 128 + wm;
  for (int i2 = 0; i2 < 2; ++i2)
    for (int j2 = 0; j2 < 2; ++j2) {
      int mb = mGlob + i2 * 16 + ((lane >> 4) << 3);
      int nn = nBase + wn + j2 * 16 + (lane & 15);
      for (int r = 0; r < 8; ++r) C[(size_t)(mb + r) * D_ + nn] = acc[i2][j2][r];
    }
}

// g_p = F0^p @ Wg[0:384] + T1 @ Wg[384:768] + T2 @ Wg[768:1152] + bg -> F1[:, colofs:colofs+128]
__global__ __launch_bounds__(256) void k_gemm_g(const float* __restrict__ F0,
    const float* __restrict__ T1, const float* __restrict__ T2,
    const float* __restrict__ Wg, const float* __restrict__ bg,
    float* __restrict__ F1, int powp, int colofs) {
  int lane = threadIdx.x & 31, wave = threadIdx.x >> 5;
  int m0 = blockIdx.x * 128 + (wave & 3) * 32;
  int n0 = blockIdx.y * 64  + (wave >> 2) * 32;
  v8f acc[2][2] = {};
  const float* srcs[3] = {F0, T1, T2};
  for (int seg = 0; seg < 3; ++seg) {
    const float* S = srcs[seg];
    const float* W = Wg + (size_t)seg * D_ * ND_;
    int p = (seg == 0) ? powp : 1;
    for (int k0 = 0; k0 < D_; k0 += 4) {
      v2f a0 = load_a_frag(S, D_, m0,      k0, lane);
      v2f a1 = load_a_frag(S, D_, m0 + 16, k0, lane);
      if (p != 1) {
        a0.x = applyPow(a0.x, p); a0.y = applyPow(a0.y, p);
        a1.x = applyPow(a1.x, p); a1.y = applyPow(a1.y, p);
      }
      v2f b0 = load_b_frag(W, ND_, k0, n0,      lane);
      v2f b1 = load_b_frag(W, ND_, k0, n0 + 16, lane);
      WMMA4(acc[0][0], a0, b0); WMMA4(acc[0][1], a0, b1);
      WMMA4(acc[1][0], a1, b0); WMMA4(acc[1][1], a1, b1);
    }
  }
  for (int i2 = 0; i2 < 2; ++i2)
    for (int j2 = 0; j2 < 2; ++j2) {
      int mb = m0 + i2 * 16 + ((lane >> 4) << 3);
      int nn = n0 + j2 * 16 + (lane & 15);
      for (int r = 0; r < 8; ++r)
        F1[(size_t)(mb + r) * D_ + colofs + nn] = acc[i2][j2][r] + bg[nn];
    }
}

__global__ void k_residual(float* __restrict__ F1, const float* __restrict__ F0) {
  size_t i = (size_t)blockIdx.x * 256 + threadIdx.x;
  F1[i] = lrelu(F1[i]) + F0[i];
}

__global__ void k_bnstats(const float* __restrict__ F1, const float* __restrict__ gamma,
                          const float* __restrict__ beta, float* __restrict__ bn) {
  __shared__ float ss[256], ss2[256];
  int c = blockIdx.x, tid = threadIdx.x;
  float s = 0.f, s2 = 0.f;
  for (int r = tid; r < BN_; r += 256) { float v = F1[(size_t)r * D_ + c]; s += v; s2 += v * v; }
  ss[tid] = s; ss2[tid] = s2; __syncthreads();
  for (int st = 128; st > 0; st >>= 1) {
    if (tid < st) { ss[tid] += ss[tid + st]; ss2[tid] += ss2[tid + st]; }
    __syncthreads();
  }
  if (tid == 0) {
    float mean = ss[0] * (1.0f / BN_);
    float var  = ss2[0] * (1.0f / BN_) - mean * mean;   // biased var, matches reference
    float scale = gamma[c] * rsqrtf(var + 1e-5f);
    bn[c]       = scale;
    bn[D_ + c]  = beta[c] - mean * scale;
  }
}

// F_final = leaky(BN(F1) @ W_F + b_F), written directly into h[b, 1+n, :]
__global__ __launch_bounds__(256) void k_gemm_final(const float* __restrict__ F1,
    const float* __restrict__ bn, const float* __restrict__ WF,
    const float* __restrict__ bF, float* __restrict__ h) {
  int lane = threadIdx.x & 31, wave = threadIdx.x >> 5;
  int m0 = blockIdx.x * 128 + (wave & 3) * 32;
  int n0 = blockIdx.y * 64  + (wave >> 2) * 32;
  const float* scale = bn;
  const float* shift = bn + D_;
  v8f acc[2][2] = {};
  for (int k0 = 0; k0 < D_; k0 += 4) {
    int kb = k0 + ((lane >> 4) << 1);
    int ra = m0 + (lane & 15);
    int rb = ra + 16;
    v2f a0, a1;
    a0.x = F1[(size_t)ra * D_ + kb]     * scale[kb]     + shift[kb];
    a0.y = F1[(size_t)ra * D_ + kb + 1] * scale[kb + 1] + shift[kb + 1];
    a1.x = F1[(size_t)rb * D_ + kb]     * scale[kb]     + shift[kb];
    a1.y = F1[(size_t)rb * D_ + kb + 1] * scale[kb + 1] + shift[kb + 1];
    v2f b0 = load_b_frag(WF, ND_, k0, n0,      lane);
    v2f b1 = load_b_frag(WF, ND_, k0, n0 + 16, lane);
    WMMA4(acc[0][0], a0, b0); WMMA4(acc[0][1], a0, b1);
    WMMA4(acc[1][0], a1, b0); WMMA4(acc[1][1], a1, b1);
  }
  for (int i2 = 0; i2 < 2; ++i2)
    for (int j2 = 0; j2 < 2; ++j2) {
      int mb = m0 + i2 * 16 + ((lane >> 4) << 3);
      int nn = n0 + j2 * 16 + (lane & 15);
      for (int r = 0; r < 8; ++r) {
        int row = mb + r;
        int bb = row >> 9, nrow = row & 511;
        h[((size_t)bb * 513 + 1 + nrow) * ND_ + nn] = lrelu(acc[i2][j2][r] + bF[nn]);
      }
    }
}

__global__ void k_depot(const float* __restrict__ depot, const float* __restrict__ Wdep,
                        const float* __restrict__ bdep, float* __restrict__ h) {
  int b = blockIdx.x, c = threadIdx.x;
  h[(size_t)b * 513 * ND_ + c] =
      depot[b * 2] * Wdep[c] + depot[b * 2 + 1] * Wdep[ND_ + c] + bdep[c];
}

__global__ void k_mean(const float* __restrict__ h, float* __restrict__ out) {
  int b = blockIdx.x, c = threadIdx.x;
  float s = 0.f;
  for (int r = 0; r < 513; ++r) s += h[((size_t)b * 513 + r) * ND_ + c];
  out[b * ND_ + c] = s * (1.0f / 513.0f);
}

// ---------------- launch ----------------

extern "C" void kernel_launch(void* const* d_in, const int* in_sizes, int n_in,
                              void* d_out, int out_size, void* d_ws, size_t ws_size,
                              hipStream_t stream) {
  const float* loc      = (const float*)d_in[0];
  const float* deadline = (const float*)d_in[1];
  const float* depot    = (const float*)d_in[3];
  const float* W_init   = (const float*)d_in[4];
  const float* b_init   = (const float*)d_in[5];
  const float* W_dep    = (const float*)d_in[6];
  const float* b_dep    = (const float*)d_in[7];
  const float* W_g[3]   = {(const float*)d_in[8], (const float*)d_in[10], (const float*)d_in[12]};
  const float* b_g[3]   = {(const float*)d_in[9], (const float*)d_in[11], (const float*)d_in[13]};
  const float* gamma    = (const float*)d_in[14];
  const float* beta     = (const float*)d_in[15];
  const float* W_F      = (const float*)d_in[16];
  const float* b_F      = (const float*)d_in[17];

  float* ws    = (float*)d_ws;
  float* stats = ws + OFF_STATS;
  float* bn    = ws + OFF_BN;
  float* X     = ws + OFF_X;
  float* L     = ws + OFF_L;
  float* F0    = ws + OFF_F0;
  float* T1    = ws + OFF_T1;
  float* T2    = ws + OFF_T2;
  float* F1    = ws + OFF_F1;
  float* h     = (float*)d_out;
  float* hmean = h + (size_t)B_ * 513 * ND_;

  k_init<<<1, 256, 0, stream>>>(deadline, stats);
  k_build_XF0<<<BN_, 128, 0, stream>>>(loc, deadline, W_init, b_init, stats, X, F0);
  k_d2min<<<32768, 256, 0, stream>>>(X, (unsigned int*)stats);
  k_build_L<<<BN_, 512, 0, stream>>>(X, stats, L);

  for (int p = 1; p <= 3; ++p) {
    const float* src = F0;
    if (p != 1) {                                   // T2 = F0^p (elementwise)
      k_pow<<<(BN_ * D_) / 256, 256, 0, stream>>>(F0, T2, p);
      src = T2;
    }
    k_gemm_L<<<dim3(4, 6, B_), 256, 0, stream>>>(L, src, T1);  // T1 = L @ F0^p
    k_gemm_L<<<dim3(4, 6, B_), 256, 0, stream>>>(L, T1, T2);   // T2 = L @ T1 (== L2 @ F0^p)
    k_gemm_g<<<dim3(128, 2, 1), 256, 0, stream>>>(F0, T1, T2, W_g[p - 1], b_g[p - 1],
                                                  F1, p, (p - 1) * ND_);
  }

  k_residual<<<(BN_ * D_) / 256, 256, 0, stream>>>(F1, F0);
  k_bnstats<<<D_, 256, 0, stream>>>(F1, gamma, beta, bn);
  k_gemm_final<<<dim3(128, 2, 1), 256, 0, stream>>>(F1, bn, W_F, b_F, h);
  k_depot<<<B_, ND_, 0, stream>>>(depot, W_dep, b_dep, h);
  k_mean<<<B_, ND_, 0, stream>>>(h, hmean);
}